// BilateralGridCP4D_48765058679256
// MI455X (gfx1250) — compile-verified
//
#include <hip/hip_runtime.h>

typedef __attribute__((ext_vector_type(2))) float v2f;
typedef __attribute__((ext_vector_type(4))) float v4f;
typedef __attribute__((ext_vector_type(8))) float v8f;

#define RANK 5

// c[r] *= lerp over LDS-resident table T (RANK x D), coordinate v in ~[-1,1],
// border clamp + align_corners=True (matches reference _interp1d).
__device__ __forceinline__ void interp_mul(const float* __restrict__ T, int D,
                                           float v, float* c) {
  float pos = (v + 1.0f) * 0.5f * (float)(D - 1);
  pos = fminf(fmaxf(pos, 0.0f), (float)(D - 1));
  int i0 = (int)pos;                 // floor (pos >= 0)
  if (i0 > D - 2) i0 = D - 2;
  float w = pos - (float)i0;
#pragma unroll
  for (int r = 0; r < RANK; ++r) {
    float a = T[r * D + i0];
    float b = T[r * D + i0 + 1];
    c[r] *= fmaf(w, b - a, a);
  }
}

__global__ __launch_bounds__(256) void bgrid_cp4d_kernel(
    const float* __restrict__ xyz, const float* __restrict__ rgb,
    const float* __restrict__ fac0, const float* __restrict__ fac1,
    const float* __restrict__ fac2, const float* __restrict__ fac3,
    const float* __restrict__ fac4, const float* __restrict__ w1,
    const float* __restrict__ b1, const float* __restrict__ w2,
    const float* __restrict__ b2, float* __restrict__ out, int N) {
  __shared__ float sF1[RANK * 8];
  __shared__ float sF2[RANK * 16];
  __shared__ float sF3[RANK * 16];
  __shared__ float sF4[RANK * 16];
  __shared__ float sCoef[8][32 * RANK];          // per-wave coef, stride 5 (conflict-free)
  __shared__ __align__(16) float sOut[8][32 * 12];  // per-wave D staging for b128 stores

  const int tid = threadIdx.x;

  // Stage the four interp tables (280 floats) into LDS.
  for (int i = tid; i < 280; i += 256) {
    if (i < 40)       sF1[i]       = fac1[i];
    else if (i < 120) sF2[i - 40]  = fac2[i - 40];
    else if (i < 200) sF3[i - 120] = fac3[i - 120];
    else              sF4[i - 200] = fac4[i - 200];
  }
  __syncthreads();

  const int lane = tid & 31;
  const int wave = tid >> 5;
  const int base = (blockIdx.x * 8 + wave) * 32;   // 32-bit indices throughout
  const int p = base + lane;
  const int pc = (p < N) ? p : (N - 1);            // clamp loads; EXEC stays full

  // ---- per-point streaming work (one-pass data -> non-temporal loads) ----
  const float x  = __builtin_nontemporal_load(xyz + pc * 3 + 0) * 0.5f;  // /BOUND
  const float y  = __builtin_nontemporal_load(xyz + pc * 3 + 1) * 0.5f;
  const float z  = __builtin_nontemporal_load(xyz + pc * 3 + 2) * 0.5f;
  const float r0 = __builtin_nontemporal_load(rgb + pc * 3 + 0);
  const float r1 = __builtin_nontemporal_load(rgb + pc * 3 + 1);
  const float r2 = __builtin_nontemporal_load(rgb + pc * 3 + 2);

  // gray = tanh(2 * (relu(rgb@W1^T + b1) @ W2^T + b2))
  float u = b2[0];
#pragma unroll
  for (int j = 0; j < 8; ++j) {
    float h = fmaf(r0, w1[3 * j + 0],
              fmaf(r1, w1[3 * j + 1],
              fmaf(r2, w1[3 * j + 2], b1[j])));
    h = fmaxf(h, 0.0f);
    u = fmaf(h, w2[j], u);
  }
  // tanh(2u) = sign(u) * (1 - 2/(exp(4|u|)+1)); overflow-safe, uses v_exp_f32
  const float e = __expf(4.0f * fabsf(u));
  float g = 1.0f - 2.0f / (e + 1.0f);
  g = copysignf(g, u);

  float c[RANK];
#pragma unroll
  for (int r = 0; r < RANK; ++r) c[r] = 1.0f;
  interp_mul(sF1, 8,  x, c);
  interp_mul(sF2, 16, y, c);
  interp_mul(sF3, 16, z, c);
  interp_mul(sF4, 16, g, c);

#pragma unroll
  for (int r = 0; r < RANK; ++r) sCoef[wave][lane * RANK + r] = c[r];
  __syncthreads();

  // ---- rank->12 projection via V_WMMA_F32_16X16X4_F32 ----
  // B = fac0^T (5x12) padded to 16 cols; K split 4 + 1. B layout mirrors A:
  // lanes 0-15 hold K=0/1, lanes 16-31 hold K=2/3, column N = lane&15.
  const bool hi = (lane >= 16);
  const int n = lane & 15;
  v2f bA; bA.x = 0.0f; bA.y = 0.0f;   // ranks 0..3
  v2f bE; bE.x = 0.0f; bE.y = 0.0f;   // rank 4 in K-slot 0
  if (n < 12) {
    const int k0 = hi ? 2 : 0;
    bA.x = fac0[n * RANK + k0];
    bA.y = fac0[n * RANK + k0 + 1];
    if (!hi) bE.x = fac0[n * RANK + 4];
  }

  const float* cw = sCoef[wave];
  float* ow = sOut[wave];
#pragma unroll
  for (int t = 0; t < 2; ++t) {  // two M=16 tiles -> 32 points per wave
    const int m = t * 16 + n;    // A row owned by this lane
    v2f aA; aA.x = cw[m * RANK + (hi ? 2 : 0)];
            aA.y = cw[m * RANK + (hi ? 3 : 1)];
    v2f aE; aE.x = hi ? 0.0f : cw[m * RANK + 4];
            aE.y = 0.0f;

    v8f d = {0.0f, 0.0f, 0.0f, 0.0f, 0.0f, 0.0f, 0.0f, 0.0f};
    d = __builtin_amdgcn_wmma_f32_16x16x4_f32(false, aA, false, bA, (short)0, d,
                                              false, false);
    d = __builtin_amdgcn_wmma_f32_16x16x4_f32(false, aE, false, bE, (short)0, d,
                                              false, false);

    // Scatter fragment -> LDS in output-linear layout (conflict-free: the two
    // 12-wide runs per VGPR land 32 banks apart).
    if (n < 12) {
      const int rbase = (t * 16 + (hi ? 8 : 0)) * 12 + n;
#pragma unroll
      for (int v = 0; v < 8; ++v) ow[rbase + v * 12] = d[v];
    }
  }
  __syncthreads();

  // ---- coalesced non-temporal b128 stores: 3 x float4 per point ----
  if (p < N) {
    const v4f* so = (const v4f*)(ow + lane * 12);
    v4f* op = (v4f*)(out + (size_t)p * 12);
#pragma unroll
    for (int k = 0; k < 3; ++k) __builtin_nontemporal_store(so[k], op + k);
  }
}

extern "C" void kernel_launch(void* const* d_in, const int* in_sizes, int n_in,
                              void* d_out, int out_size, void* d_ws, size_t ws_size,
                              hipStream_t stream) {
  const float* xyz  = (const float*)d_in[0];
  const float* rgb  = (const float*)d_in[1];
  const float* fac0 = (const float*)d_in[2];
  const float* fac1 = (const float*)d_in[3];
  const float* fac2 = (const float*)d_in[4];
  const float* fac3 = (const float*)d_in[5];
  const float* fac4 = (const float*)d_in[6];
  const float* w1   = (const float*)d_in[7];
  const float* b1   = (const float*)d_in[8];
  const float* w2   = (const float*)d_in[9];
  const float* b2   = (const float*)d_in[10];
  const int N = in_sizes[0] / 3;
  const int blocks = (N + 255) / 256;  // 8 waves/block, 32 points/wave
  bgrid_cp4d_kernel<<<blocks, 256, 0, stream>>>(xyz, rgb, fac0, fac1, fac2,
                                                fac3, fac4, w1, b1, w2, b2,
                                                (float*)d_out, N);
}